// TripleCrossMessageBlock_40475771797590
// MI455X (gfx1250) — compile-verified
//
#include <hip/hip_runtime.h>
#include <math.h>

typedef __attribute__((ext_vector_type(2))) float v2f;
typedef __attribute__((ext_vector_type(8))) float v8f;

static constexpr int N_NODES = 10000;
static constexpr int N_EDGES = 160000;
static constexpr int F       = 128;
static constexpr int F8      = 8 * F;     // 1024
static constexpr int NRB     = 32;

// ---------------------------------------------------------------------------
// Stage 1a: h = silu(s @ W1 + b1)       [N_NODES x F]
// One block = 16 node rows; 8 waves, wave w -> cols [16w, 16w+16).
// ---------------------------------------------------------------------------
__global__ void __launch_bounds__(256) phi1_kernel(const float* __restrict__ s,
                                                   const float* __restrict__ W1,
                                                   const float* __restrict__ b1,
                                                   float* __restrict__ h) {
    const int wave = threadIdx.x >> 5;
    const int lane = threadIdx.x & 31;
    const int half = lane >> 4;       // 0: K+{0,1}, 1: K+{2,3}
    const int n    = lane & 15;       // N within tile / M row for A loads
    const int rowA = blockIdx.x * 16 + n;          // A-matrix row (M = lane&15)
    const int col  = wave * 16 + n;                // output column

    v8f acc = {};
#pragma unroll
    for (int k = 0; k < F / 4; ++k) {
        const int kk = 4 * k + 2 * half;
        v2f a, b;
        a.x = s[rowA * F + kk];
        a.y = s[rowA * F + kk + 1];
        b.x = W1[kk * F + col];
        b.y = W1[(kk + 1) * F + col];
        acc = __builtin_amdgcn_wmma_f32_16x16x4_f32(false, a, false, b,
                                                    (short)0, acc, false, false);
    }
    const float bias = b1[col];
#pragma unroll
    for (int r = 0; r < 8; ++r) {
        const int m = r + 8 * half;                // C/D layout: M = r + 8*half
        float val = acc[r] + bias;
        val = val / (1.0f + __expf(-val));         // SiLU
        h[(blockIdx.x * 16 + m) * F + col] = val;
    }
}

// ---------------------------------------------------------------------------
// Stage 1b: phi = h @ W2 + b2           [N_NODES x 8F]
// grid (625, 8): blockIdx.y selects 128-col group, wave w -> 16 cols within.
// ---------------------------------------------------------------------------
__global__ void __launch_bounds__(256) phi2_kernel(const float* __restrict__ h,
                                                   const float* __restrict__ W2,
                                                   const float* __restrict__ b2,
                                                   float* __restrict__ phi) {
    const int wave = threadIdx.x >> 5;
    const int lane = threadIdx.x & 31;
    const int half = lane >> 4;
    const int n    = lane & 15;
    const int rowA = blockIdx.x * 16 + n;
    const int col  = blockIdx.y * 128 + wave * 16 + n;

    v8f acc = {};
#pragma unroll
    for (int k = 0; k < F / 4; ++k) {
        const int kk = 4 * k + 2 * half;
        v2f a, b;
        a.x = h[rowA * F + kk];
        a.y = h[rowA * F + kk + 1];
        b.x = W2[kk * F8 + col];
        b.y = W2[(kk + 1) * F8 + col];
        acc = __builtin_amdgcn_wmma_f32_16x16x4_f32(false, a, false, b,
                                                    (short)0, acc, false, false);
    }
    const float bias = b2[col];
#pragma unroll
    for (int r = 0; r < 8; ++r) {
        const int m = r + 8 * half;
        phi[(blockIdx.x * 16 + m) * F8 + col] = acc[r] + bias;
    }
}

// ---------------------------------------------------------------------------
// Stage 2: fused edge kernel.
// Block = one tile of 16 edges, 8 waves; wave w owns features [16w, 16w+16).
//   W  = (f1*r1 + f2*r2 + f3*r3) @ Wr + br*(f1+f2+f3)   via WMMA (K=32)
//   x  = phi[idx_j] * W ; split into 8 chunks; scalar + vector recombination;
//   atomic scatter-add into out_s / out_v (pre-initialized with s / v).
// ---------------------------------------------------------------------------
__global__ void __launch_bounds__(256) edge_kernel(
        const float* __restrict__ phi, const float* __restrict__ v,
        const float* __restrict__ re1, const float* __restrict__ re2,
        const float* __restrict__ re3,
        const float* __restrict__ f1, const float* __restrict__ f2,
        const float* __restrict__ f3,
        const float* __restrict__ u1, const float* __restrict__ u2,
        const float* __restrict__ u3,
        const long long* __restrict__ edge_index,
        const float* __restrict__ Wr, const float* __restrict__ br,
        float* __restrict__ out_s, float* __restrict__ out_v) {
    __shared__ float s_f[3][16];     // per-edge cutoffs
    __shared__ float s_fsum[16];
    __shared__ int   s_idx_i[16], s_idx_j[16];
    __shared__ float s_u[3][3][16];  // [branch][xyz][edge]

    const int tid  = threadIdx.x;
    const int tile = blockIdx.x;     // 16 edges per tile

    if (tid < 16) {
        const int e = tile * 16 + tid;
        const float a = f1[e], b = f2[e], c = f3[e];
        s_f[0][tid] = a; s_f[1][tid] = b; s_f[2][tid] = c;
        s_fsum[tid] = a + b + c;
        s_idx_i[tid] = (int)edge_index[e];            // row 0
        s_idx_j[tid] = (int)edge_index[N_EDGES + e];  // row 1
    } else if (tid < 160) {
        const int t  = tid - 16;     // 0..143
        const int br_ = t / 48;      // branch
        const int cp  = (t / 16) % 3;
        const int m   = t % 16;
        const int e   = tile * 16 + m;
        const float* u = (br_ == 0) ? u1 : (br_ == 1) ? u2 : u3;
        s_u[br_][cp][m] = u[e * 3 + cp];
    }
    __syncthreads();

    const int wave = tid >> 5;
    const int lane = tid & 31;
    const int half = lane >> 4;
    const int n    = lane & 15;
    const int eA   = tile * 16 + n;          // A-matrix row (edge), per lane
    const float fa = s_f[0][n], fb = s_f[1][n], fc = s_f[2][n];

    // Build combined radial A tile [16 x 32] in registers (8 K-chunks of 4).
    v2f A[8];
#pragma unroll
    for (int k = 0; k < 8; ++k) {
        const int c0 = 4 * k + 2 * half;
        A[k].x = fa * re1[eA * NRB + c0]     + fb * re2[eA * NRB + c0]     + fc * re3[eA * NRB + c0];
        A[k].y = fa * re1[eA * NRB + c0 + 1] + fb * re2[eA * NRB + c0 + 1] + fc * re3[eA * NRB + c0 + 1];
    }

    // 8 output chunks, each a 16x16 tile at column base c*128 + wave*16.
    v8f acc[8];
#pragma unroll
    for (int c = 0; c < 8; ++c) {
        const int col = c * 128 + wave * 16 + n;
        v8f a8 = {};
#pragma unroll
        for (int k = 0; k < 8; ++k) {
            const int kk = 4 * k + 2 * half;
            v2f b;
            b.x = Wr[kk * F8 + col];
            b.y = Wr[(kk + 1) * F8 + col];
            a8 = __builtin_amdgcn_wmma_f32_16x16x4_f32(false, A[k], false, b,
                                                       (short)0, a8, false, false);
        }
        acc[c] = a8;
    }

    const int f = wave * 16 + n;             // feature column 0..127
#pragma unroll
    for (int r = 0; r < 8; ++r) {
        const int m    = r + 8 * half;       // edge within tile
        const int i    = s_idx_i[m];
        const int j    = s_idx_j[m];
        const float fs = s_fsum[m];

        float x[8];
#pragma unroll
        for (int c = 0; c < 8; ++c) {
            const int col = c * 128 + f;
            const float w = acc[c][r] + br[col] * fs;
            x[c] = phi[j * F8 + col] * w;
        }
        // chunks: 0=x_s 1=x_vv 2..4=x_vs1..3 5..7=x_vc1..3
        const float vjx = v[j * 384 + f];
        const float vjy = v[j * 384 + 128 + f];
        const float vjz = v[j * 384 + 256 + f];

        const float u1x = s_u[0][0][m], u1y = s_u[0][1][m], u1z = s_u[0][2][m];
        const float u2x = s_u[1][0][m], u2y = s_u[1][1][m], u2z = s_u[1][2][m];
        const float u3x = s_u[2][0][m], u3y = s_u[2][1][m], u3z = s_u[2][2][m];

        // cross(vj, u) along xyz axis
        const float c1x = vjy * u1z - vjz * u1y, c1y = vjz * u1x - vjx * u1z, c1z = vjx * u1y - vjy * u1x;
        const float c2x = vjy * u2z - vjz * u2y, c2y = vjz * u2x - vjx * u2z, c2z = vjx * u2y - vjy * u2x;
        const float c3x = vjy * u3z - vjz * u3y, c3y = vjz * u3x - vjx * u3z, c3z = vjx * u3y - vjy * u3x;

        const float xvx = vjx * x[1] + x[2] * u1x + x[3] * u2x + x[4] * u3x
                        + x[5] * c1x + x[6] * c2x + x[7] * c3x;
        const float xvy = vjy * x[1] + x[2] * u1y + x[3] * u2y + x[4] * u3y
                        + x[5] * c1y + x[6] * c2y + x[7] * c3y;
        const float xvz = vjz * x[1] + x[2] * u1z + x[3] * u2z + x[4] * u3z
                        + x[5] * c1z + x[6] * c2z + x[7] * c3z;

        atomicAdd(&out_s[i * F + f], x[0]);
        atomicAdd(&out_v[i * 384 + f],        xvx);
        atomicAdd(&out_v[i * 384 + 128 + f],  xvy);
        atomicAdd(&out_v[i * 384 + 256 + f],  xvz);
    }
}

// ---------------------------------------------------------------------------
extern "C" void kernel_launch(void* const* d_in, const int* in_sizes, int n_in,
                              void* d_out, int out_size, void* d_ws, size_t ws_size,
                              hipStream_t stream) {
    const float*     s    = (const float*)d_in[0];
    const float*     v    = (const float*)d_in[1];
    const float*     re1  = (const float*)d_in[2];
    const float*     re2  = (const float*)d_in[3];
    const float*     re3  = (const float*)d_in[4];
    const float*     f1   = (const float*)d_in[5];
    const float*     f2   = (const float*)d_in[6];
    const float*     f3   = (const float*)d_in[7];
    const float*     u1   = (const float*)d_in[8];
    const float*     u2   = (const float*)d_in[9];
    const float*     u3   = (const float*)d_in[10];
    const long long* eidx = (const long long*)d_in[11];
    const float*     W1   = (const float*)d_in[12];
    const float*     b1   = (const float*)d_in[13];
    const float*     W2   = (const float*)d_in[14];
    const float*     b2   = (const float*)d_in[15];
    const float*     Wr   = (const float*)d_in[16];
    const float*     br   = (const float*)d_in[17];

    float* h   = (float*)d_ws;                       // [N_NODES x F]
    float* phi = h + (size_t)N_NODES * F;            // [N_NODES x 8F]

    float* out_s = (float*)d_out;                    // [N_NODES x F]
    float* out_v = out_s + (size_t)N_NODES * F;      // [N_NODES x 3 x F]

    // Node-side MLP (WMMA tiled).
    phi1_kernel<<<N_NODES / 16, 256, 0, stream>>>(s, W1, b1, h);
    phi2_kernel<<<dim3(N_NODES / 16, 8), 256, 0, stream>>>(h, W2, b2, phi);

    // Initialize outputs with residual (s, v); edge kernel accumulates on top.
    hipMemcpyAsync(out_s, s, (size_t)N_NODES * F * sizeof(float),
                   hipMemcpyDeviceToDevice, stream);
    hipMemcpyAsync(out_v, v, (size_t)N_NODES * 3 * F * sizeof(float),
                   hipMemcpyDeviceToDevice, stream);

    // Fused radial-filter GEMM + gather + cross recombination + scatter-add.
    edge_kernel<<<N_EDGES / 16, 256, 0, stream>>>(
        phi, v, re1, re2, re3, f1, f2, f3, u1, u2, u3, eidx, Wr, br,
        out_s, out_v);
}